// FRCLoss_77575699300877
// MI455X (gfx1250) — compile-verified
//
#include <hip/hip_runtime.h>
#include <math.h>

// Problem geometry (fixed by the reference)
#define NCLS   19
#define NBATCH 8
#define NCH    64
#define HW     (512 * 512)
#define BPB    64                 // blocks per batch sample
#define PPB    (HW / BPB)         // pixels per block = 4096
#define NBLK   (NBATCH * BPB)     // 512 partial-accumulator blocks
#define STR    65                 // LDS row stride (bank-conflict padding)
#define KDIM   18                 // classes 1..18 form the embedding matrix

typedef __attribute__((ext_vector_type(2))) float v2f;
typedef __attribute__((ext_vector_type(8))) float v8f;

// ---------------------------------------------------------------------------
// Kernel 1: bandwidth-bound segmented reduction.
// Each block: 4096 pixels of one batch sample, all 64 channels, float4 loads.
// LDS histogram sums[19][64] (stride 65) + counts[19], via ds_add_f32 atomics.
// Writes per-block partials to workspace (deterministic final reduction later).
// ---------------------------------------------------------------------------
__global__ __launch_bounds__(256) void frc_accum(const float* __restrict__ feat,
                                                 const int* __restrict__ label,
                                                 float* __restrict__ psums,
                                                 float* __restrict__ pcnts) {
    __shared__ float s_sum[NCLS * STR];
    __shared__ float s_cnt[NCLS];
    const int t = threadIdx.x;
    const int b = blockIdx.y;

    for (int i = t; i < NCLS * STR; i += 256) s_sum[i] = 0.f;
    if (t < NCLS) s_cnt[t] = 0.f;
    __syncthreads();

    const int    pix0  = blockIdx.x * PPB;
    const int*   labb  = label + (size_t)b * HW;
    const float* featb = feat + (size_t)b * NCH * HW;

    for (int base = 0; base < PPB; base += 256 * 4) {
        const int p = pix0 + base + t * 4;

        // next chunk's labels: speculative prefetch (global_prefetch path)
        if (base + 256 * 4 < PPB) __builtin_prefetch(labb + p + 256 * 4, 0, 1);

        int4 l4 = *(const int4*)(labb + p);
        int c0 = (l4.x == 255) ? 0 : l4.x;
        int c1 = (l4.y == 255) ? 0 : l4.y;
        int c2 = (l4.z == 255) ? 0 : l4.z;
        int c3 = (l4.w == 255) ? 0 : l4.w;

        atomicAdd(&s_cnt[c0], 1.f);
        atomicAdd(&s_cnt[c1], 1.f);
        atomicAdd(&s_cnt[c2], 1.f);
        atomicAdd(&s_cnt[c3], 1.f);

#pragma unroll 8
        for (int d = 0; d < NCH; d++) {
            float4 f4 = *(const float4*)(featb + (size_t)d * HW + p);
            atomicAdd(&s_sum[c0 * STR + d], f4.x);
            atomicAdd(&s_sum[c1 * STR + d], f4.y);
            atomicAdd(&s_sum[c2 * STR + d], f4.z);
            atomicAdd(&s_sum[c3 * STR + d], f4.w);
        }
    }
    __syncthreads();

    const int blk = b * BPB + blockIdx.x;
    float* out_s = psums + (size_t)blk * NCLS * NCH;
    for (int i = t; i < NCLS * NCH; i += 256)
        out_s[i] = s_sum[(i / NCH) * STR + (i % NCH)];
    if (t < NCLS) pcnts[blk * NCLS + t] = s_cnt[t];
}

// ---------------------------------------------------------------------------
// Kernel 2: deterministic reduction of block partials -> embeddings E[18][64].
//   means[b,c,d] = sum/(count+1e-6); E[c-1] = (sum_b means) / num[c]
// ---------------------------------------------------------------------------
__global__ void frc_reduce(const float* __restrict__ psums,
                           const float* __restrict__ pcnts,
                           float* __restrict__ E) {
    const int idx = blockIdx.x * 256 + threadIdx.x;
    if (idx >= KDIM * NCH) return;
    const int c = idx / NCH + 1;   // classes 1..18
    const int d = idx % NCH;

    float acc = 0.f, num = 0.f;
    for (int b = 0; b < NBATCH; b++) {
        float s = 0.f, cnt = 0.f;
        for (int k = 0; k < BPB; k++) {
            const int blk = b * BPB + k;
            s   += psums[((size_t)blk * NCLS + c) * NCH + d];
            cnt += pcnts[blk * NCLS + c];
        }
        acc += s / (cnt + 1e-6f);
        if (cnt > 0.f) num += 1.f;
    }
    E[(c - 1) * NCH + d] = acc / num;
}

// ---------------------------------------------------------------------------
// Kernel 3: single wave32. Gram = E * E^T via V_WMMA_F32_16X16X4_F32
// (E padded to 32x64; 2x2 tiles of 16x16, K=64 -> 16 WMMA steps per tile).
// A fragment (16x4): lane L holds rows M=L%16, K = kb + 2*(L/16) + v.
// B = E^T has the identical per-lane load pattern (lane = column of B = row of E).
// Then LDS scatter per the 16x16 f32 C/D layout, normalize, log-softmax, loss.
// ---------------------------------------------------------------------------
__global__ __launch_bounds__(32) void frc_gram_loss(const float* __restrict__ E,
                                                    float* __restrict__ out) {
    __shared__ float dots[KDIM][KDIM + 1];
    __shared__ float sq[KDIM];
    __shared__ float rowloss[KDIM];

    const int lane = threadIdx.x;        // 0..31, EXEC all ones for WMMA
    const int m    = lane & 15;
    const int kh   = (lane >> 4) * 2;
    const int r1   = 16 + m;             // row-tile-1 row (only 16,17 valid)

    v8f acc00 = {}; v8f acc01 = {}; v8f acc10 = {}; v8f acc11 = {};

    for (int kb = 0; kb < NCH; kb += 4) {
        v2f a0, a1;
#pragma unroll
        for (int v = 0; v < 2; v++) {
            const int k = kb + kh + v;
            a0[v] = E[m * NCH + k];                          // rows 0..15
            a1[v] = (r1 < KDIM) ? E[r1 * NCH + k] : 0.f;     // rows 16..31 (pad 0)
        }
        acc00 = __builtin_amdgcn_wmma_f32_16x16x4_f32(false, a0, false, a0,
                                                      (short)0, acc00, false, false);
        acc01 = __builtin_amdgcn_wmma_f32_16x16x4_f32(false, a0, false, a1,
                                                      (short)0, acc01, false, false);
        acc10 = __builtin_amdgcn_wmma_f32_16x16x4_f32(false, a1, false, a0,
                                                      (short)0, acc10, false, false);
        acc11 = __builtin_amdgcn_wmma_f32_16x16x4_f32(false, a1, false, a1,
                                                      (short)0, acc11, false, false);
    }

    // C/D layout: VGPR r -> M = (lane<16 ? r : 8+r), N = lane%16
    const int col   = lane & 15;
    const int rbase = (lane >> 4) * 8;
#pragma unroll
    for (int r = 0; r < 8; r++) {
        const int row = rbase + r;
        if (row < KDIM && col < KDIM)             dots[row][col]           = acc00[r];
        if (row < KDIM && (16 + col) < KDIM)      dots[row][16 + col]      = acc01[r];
        if ((16 + row) < KDIM && col < KDIM)      dots[16 + row][col]      = acc10[r];
        if ((16 + row) < KDIM && (16 + col) < KDIM) dots[16 + row][16 + col] = acc11[r];
    }
    __syncthreads();

    if (lane < KDIM) sq[lane] = dots[lane][lane];   // squared norms (Gram diagonal)
    __syncthreads();

    if (lane < KDIM) {
        const float si = sq[lane];
        float c[KDIM];
        float mx = -1e30f;
#pragma unroll
        for (int j = 0; j < KDIM; j++) {
            c[j] = dots[lane][j] / (si * sq[j]);
            mx = fmaxf(mx, c[j]);
        }
        float se = 0.f;
#pragma unroll
        for (int j = 0; j < KDIM; j++) se += __expf(c[j] - mx);
        rowloss[lane] = -(c[lane] - mx - __logf(se));
    }
    __syncthreads();

    if (lane == 0) {
        float s = 0.f;
#pragma unroll
        for (int j = 0; j < KDIM; j++) s += rowloss[j];
        out[0] = s / (float)KDIM;
    }
}

// ---------------------------------------------------------------------------
extern "C" void kernel_launch(void* const* d_in, const int* in_sizes, int n_in,
                              void* d_out, int out_size, void* d_ws, size_t ws_size,
                              hipStream_t stream) {
    const float* feat  = (const float*)d_in[0];   // [8, 64, 512, 512] fp32
    const int*   label = (const int*)d_in[1];     // [8, 1, 512, 512] int32
    float*       out   = (float*)d_out;           // scalar loss

    // Workspace layout (floats):
    //   psums : NBLK * 19 * 64   (per-block class/channel partial sums)
    //   pcnts : NBLK * 19        (per-block class pixel counts)
    //   E     : 18 * 64          (embeddings)
    float* psums = (float*)d_ws;
    float* pcnts = psums + (size_t)NBLK * NCLS * NCH;
    float* E     = pcnts + (size_t)NBLK * NCLS;

    dim3 g1(BPB, NBATCH);
    frc_accum<<<g1, 256, 0, stream>>>(feat, label, psums, pcnts);

    frc_reduce<<<(KDIM * NCH + 255) / 256, 256, 0, stream>>>(psums, pcnts, E);

    frc_gram_loss<<<1, 32, 0, stream>>>(E, out);
}